// VectorQuantizer_70171175682421
// MI455X (gfx1250) — compile-verified
//
#include <hip/hip_runtime.h>
#include <hip/hip_bf16.h>

typedef __attribute__((ext_vector_type(2))) float v2f;
typedef __attribute__((ext_vector_type(8))) float v8f;

#define B_  32
#define D_  64
#define T_  2048
#define K_  1024
#define N_  (B_*T_)          // 65536
#define EPSN 1e-12f

// d_out layout (floats): [loss][quantized B*D*T][indices N][n_resurrected]
#define OUT_LOSS   0
#define OUT_Q      1
#define OUT_IDX    (1 + N_*D_)
#define OUT_NRES   (1 + N_*D_ + N_)

// workspace layout (floats)
#define WS_WNORM 0
#define WS_PART  (K_*D_)

// ---------------- Kernel 1: L2-normalize codebook rows ----------------
__global__ void vq_norm_weight(const float* __restrict__ w, float* __restrict__ wn) {
    int wave = threadIdx.x >> 5;
    int lane = threadIdx.x & 31;
    int row  = blockIdx.x * 8 + wave;           // 128 blocks * 8 waves = 1024 rows
    const float* src = w + row * D_ + lane * 2;
    float a = src[0], b = src[1];
    float s = a * a + b * b;
    #pragma unroll
    for (int off = 16; off >= 1; off >>= 1)
        s += __shfl_xor(s, off, 32);
    float scale = 1.0f / fmaxf(sqrtf(s), EPSN);
    float* dst = wn + row * D_ + lane * 2;
    dst[0] = a * scale;
    dst[1] = b * scale;
}

// ---------------- Kernel 2: main VQ (WMMA f32 dots + argmax + gather) ----------------
#define CBS 66                               // padded LDS row stride for codebook
#define XSS 66                               // padded LDS row stride for x tile
#define CB_OFF   0
#define XS_OFF   (K_*CBS)                    // 67584
#define CV_OFF   (XS_OFF + 64*XSS)           // candidate values [16][64]
#define CI_OFF   (CV_OFF + 1024)             // candidate indices [16][64]
#define IDXF_OFF (CI_OFF + 1024)             // final indices [64]
#define RED_OFF  (IDXF_OFF + 64)             // loss reduction [128]
#define SMEM_FLOATS (RED_OFF + 128)          // 74048 floats = 296192 B (< 320 KB WGP LDS)

__launch_bounds__(128, 1)
__global__ void vq_main(const float* __restrict__ x, const float* __restrict__ wn,
                        float* __restrict__ out, float* __restrict__ partials) {
    extern __shared__ float smem[];
    float* cb   = smem + CB_OFF;
    float* xs   = smem + XS_OFF;
    float* cv   = smem + CV_OFF;
    int*   ci   = (int*)(smem + CI_OFF);
    int*   idxs = (int*)(smem + IDXF_OFF);
    float* red  = smem + RED_OFF;

    const int tid = threadIdx.x;
    const int n0  = blockIdx.x * 64;         // 1024 blocks * 64 rows
    const int b   = n0 / T_;                 // 64 | 2048 -> tile never straddles b
    const int t0  = n0 % T_;

    // Stage full normalized codebook into LDS (coalesced global reads)
    for (int i = tid; i < K_ * D_; i += 128) {
        int r = i >> 6, d = i & 63;
        cb[r * CBS + d] = wn[i];
    }
    // Stage x tile: xs[r][d] = x[b, d, t0+r]  (coalesced: consecutive tid -> consecutive t)
    const float* xb = x + (size_t)b * D_ * T_ + t0;
    for (int i = tid; i < 64 * 64; i += 128) {
        int d = i >> 6, r = i & 63;
        xs[r * XSS + d] = xb[(size_t)d * T_ + r];
    }
    __syncthreads();

    // Row-wise L2 normalize the x tile
    if (tid < 64) {
        float s = 0.f;
        #pragma unroll 8
        for (int d = 0; d < 64; ++d) { float v = xs[tid * XSS + d]; s += v * v; }
        float sc = 1.0f / fmaxf(sqrtf(s), EPSN);
        #pragma unroll 8
        for (int d = 0; d < 64; ++d) xs[tid * XSS + d] *= sc;
    }
    __syncthreads();

    const int wave = tid >> 5;
    const int lane = tid & 31;
    const int l15  = lane & 15;
    const int hi   = lane >> 4;              // lane half: selects K offset in A/B frags

    for (int rg = 0; rg < 4; ++rg) {
        // A fragments (16x4 f32 per step): lane<16 holds row l15 k={4j,4j+1},
        // lane>=16 holds row l15 k={4j+2,4j+3}  (ISA 7.12.2 layout)
        v2f afrag[16];
        const float* arow = xs + (rg * 16 + l15) * XSS + 2 * hi;
        #pragma unroll
        for (int j = 0; j < 16; ++j) {
            afrag[j].x = arow[4 * j];
            afrag[j].y = arow[4 * j + 1];
        }

        float bestv[8];
        int   besti[8];
        #pragma unroll
        for (int r = 0; r < 8; ++r) { bestv[r] = -3.4e38f; besti[r] = 0; }

        for (int ct = 0; ct < 16; ++ct) {    // 16 column tiles of 16 -> 256 cols per wave
            int mycol = wave * 256 + ct * 16 + l15;
            const float* brow = cb + mycol * CBS + 2 * hi;  // B frag: lane = column
            v8f c = {0.f, 0.f, 0.f, 0.f, 0.f, 0.f, 0.f, 0.f};
            #pragma unroll
            for (int j = 0; j < 16; ++j) {   // accumulate over D=64, 4 per WMMA
                v2f bf;
                bf.x = brow[4 * j];
                bf.y = brow[4 * j + 1];
                c = __builtin_amdgcn_wmma_f32_16x16x4_f32(
                        false, afrag[j], false, bf, (short)0, c, false, false);
            }
            // C layout: VGPR r -> row rg*16 + r + 8*hi, col = mycol
            #pragma unroll
            for (int r = 0; r < 8; ++r) {
                float v = c[r];
                if (v > bestv[r]) { bestv[r] = v; besti[r] = mycol; } // strict > : first max
            }
        }
        // Merge candidates across lanes/waves with first-index tie-break
        #pragma unroll
        for (int r = 0; r < 8; ++r) {
            int rl = r + 8 * hi;
            int slot = wave * 16 + l15;
            cv[rl * 64 + slot] = bestv[r];
            ci[rl * 64 + slot] = besti[r];
        }
        __syncthreads();
        if (tid < 16) {
            float bv = -3.4e38f; int bi = 0x7fffffff;
            for (int s = 0; s < 64; ++s) {
                float v = cv[tid * 64 + s]; int i = ci[tid * 64 + s];
                if (v > bv || (v == bv && i < bi)) { bv = v; bi = i; }
            }
            int rglob = rg * 16 + tid;
            idxs[rglob] = bi;
            out[OUT_IDX + n0 + rglob] = (float)bi;   // indices as float
        }
        __syncthreads();
    }

    // Gather quantized rows, write transposed [B,D,T], accumulate loss partial
    float lsum = 0.f;
    {
        int r = tid & 63;
        int dbase = (tid >> 6) * 32;
        int qidx = idxs[r];
        const float* qrow = cb + qidx * CBS;
        float* orow = out + OUT_Q + (size_t)b * D_ * T_ + t0 + r;
        #pragma unroll 4
        for (int dd = 0; dd < 32; ++dd) {
            int d = dbase + dd;
            float q  = qrow[d];
            float xv = xs[r * XSS + d];
            float df = q - xv;
            lsum += df * df;
            orow[(size_t)d * T_] = q;        // fixed d: lanes -> consecutive t (coalesced)
        }
    }
    red[tid] = lsum;
    __syncthreads();
    for (int s2 = 64; s2 > 0; s2 >>= 1) {
        if (tid < s2) red[tid] += red[tid + s2];
        __syncthreads();
    }
    if (tid == 0) partials[blockIdx.x] = red[0];
}

// ---------------- Kernel 3: deterministic loss reduction + tail ----------------
__global__ void vq_finish(const float* __restrict__ partials, float* __restrict__ out) {
    __shared__ float red[256];
    int tid = threadIdx.x;
    float s = 0.f;
    for (int i = tid; i < 1024; i += 256) s += partials[i];
    red[tid] = s;
    __syncthreads();
    for (int k = 128; k > 0; k >>= 1) {
        if (tid < k) red[tid] += red[tid + k];
        __syncthreads();
    }
    if (tid == 0) {
        // loss = q_latent + 0.25*e_latent = 1.25 * mean((q - x_norm)^2)
        out[OUT_LOSS] = 1.25f * red[0] / (float)((size_t)N_ * D_);
        out[OUT_NRES] = 0.0f;
    }
}

extern "C" void kernel_launch(void* const* d_in, const int* in_sizes, int n_in,
                              void* d_out, int out_size, void* d_ws, size_t ws_size,
                              hipStream_t stream) {
    (void)in_sizes; (void)n_in; (void)out_size; (void)ws_size;
    const float* x = (const float*)d_in[0];   // [32, 64, 2048] f32
    const float* w = (const float*)d_in[1];   // [1024, 64] f32
    float* out = (float*)d_out;
    float* ws  = (float*)d_ws;
    float* wnorm    = ws + WS_WNORM;          // 256 KB
    float* partials = ws + WS_PART;           // 4 KB

    vq_norm_weight<<<K_ / 8, 256, 0, stream>>>(w, wnorm);
    vq_main<<<N_ / 64, 128, SMEM_FLOATS * sizeof(float), stream>>>(x, wnorm, out, partials);
    vq_finish<<<1, 256, 0, stream>>>(partials, out);
}